// SGCModel_6700148982288
// MI455X (gfx1250) — compile-verified
//
#include <hip/hip_runtime.h>
#include <hip/hip_bf16.h>

// ---------------------------------------------------------------------------
// SGC (K=2) on MI455X / gfx1250.
//   deg -> rsqrt -> 2x (self-loop-init + edge scatter-add SpMM)
//   -> FP32 WMMA GEMM [N,64]x[64,40]+b -> in-place wave32 log-softmax.
// Problem constants fixed by the reference: D=64, C=40.
// ---------------------------------------------------------------------------

typedef __attribute__((ext_vector_type(2))) float v2f;
typedef __attribute__((ext_vector_type(8))) float v8f;

constexpr int DD = 64;   // feature dim
constexpr int CD = 40;   // classes

// ---------------- degree / norm ----------------

__global__ void __launch_bounds__(256)
init_deg_kernel(float* __restrict__ deg, int n) {
  int i = blockIdx.x * 256 + threadIdx.x;
  if (i < n) deg[i] = 1.0f;                 // self-loop contributes 1
}

__global__ void __launch_bounds__(256)
deg_edges_kernel(const long long* __restrict__ ei, float* __restrict__ deg, int e_cnt) {
  int e = blockIdx.x * 256 + threadIdx.x;
  if (e < e_cnt) {
    long long r = ei[e];                    // row = edge_index[0][e]
    unsafeAtomicAdd(&deg[r], 1.0f);         // -> global_atomic_add_f32
  }
}

__global__ void __launch_bounds__(256)
rsqrt_inplace_kernel(float* __restrict__ deg, int n) {
  int i = blockIdx.x * 256 + threadIdx.x;
  if (i < n) deg[i] = rsqrtf(deg[i]);       // deg >= 1 always (self loops)
}

// ---------------- SpMM hop ----------------

// x_out[i] = dis[i]^2 * x_in[i]  (self-loop term doubles as the zero-init)
__global__ void __launch_bounds__(256)
self_init_kernel(const float* __restrict__ xin, const float* __restrict__ dis,
                 float* __restrict__ xout, int n) {
  int t = blockIdx.x * 256 + threadIdx.x;   // one float4 (D=64 -> 16 per node)
  if (t >= n * 16) return;
  int i = t >> 4;
  float s = dis[i];
  s = s * s;
  float4 v = ((const float4*)xin)[t];
  float4 o;
  o.x = s * v.x; o.y = s * v.y; o.z = s * v.z; o.w = s * v.w;
  ((float4*)xout)[t] = o;
}

// One wave32 per edge; lane l owns features [2l, 2l+1].
__global__ void __launch_bounds__(256)
spmm_edges_kernel(const long long* __restrict__ ei, const float* __restrict__ dis,
                  const float* __restrict__ xin, float* __restrict__ xout, int e_cnt) {
  int e = blockIdx.x * 8 + (int)(threadIdx.x >> 5);
  if (e >= e_cnt) return;
  int lane = threadIdx.x & 31;

  // speculative prefetch of the streaming edge-index arrays (HBM-resident)
  int ep = e + 4096;
  if (ep < e_cnt) {
    __builtin_prefetch(&ei[ep], 0, 1);                  // global_prefetch_b8
    __builtin_prefetch(&ei[(size_t)e_cnt + ep], 0, 1);
  }

  long long r = ei[e];
  long long c = ei[(size_t)e_cnt + e];
  float norm = dis[r] * dis[c];
  const float2 v = *(const float2*)(xin + (size_t)c * DD + lane * 2);
  float* dst = xout + (size_t)r * DD + lane * 2;
  unsafeAtomicAdd(dst + 0, norm * v.x);     // coalesced atomic scatter, L2-resident
  unsafeAtomicAdd(dst + 1, norm * v.y);
}

// ---------------- FP32 WMMA GEMM:  out[N,40] = X[N,64] * W[64,40] + bias ----------------
// One wave per (16-row tile, 16-col tile); 3 waves / block cover C=40 (padded to 48).
// A 16x4 f32 frag: lane = M (mod 16), VGPR0 = K0 (lanes 0-15) / K2 (lanes 16-31), VGPR1 = K1/K3.
// B 4x16 f32 frag mirrors A with lane = N.
// Padded B columns (40..47) are loaded with a clamped in-bounds index: they only
// pollute D columns we never store, so no masking is needed.

__global__ void __launch_bounds__(96)
gemm_wmma_kernel(const float* __restrict__ X, const float* __restrict__ Wm,
                 const float* __restrict__ bias, float* __restrict__ out, int n) {
  const int wave = (int)(threadIdx.x >> 5);       // 0..2 -> col tile
  const int lane = (int)(threadIdx.x & 31);
  const int row0 = blockIdx.x * 16;
  const int c0   = wave * 16;
  const int m    = lane & 15;
  const int kk   = (lane >> 4) << 1;              // 0 (lanes 0-15) or 2 (lanes 16-31)

  // clamp row (garbage rows only pollute rows we never store)
  int rA = row0 + m;
  if (rA >= n) rA = n - 1;
  const float* xrow = X + (size_t)rA * DD + kk;   // immediate offsets k, k+1

  const int cW  = c0 + m;
  const int cWc = (cW < CD) ? cW : (CD - 1);      // clamp, no mask needed
  const float* wcol = Wm + (size_t)kk * CD + cWc; // immediate offsets k*CD

  v8f acc = {};
#pragma unroll
  for (int k = 0; k < DD; k += 4) {
    v2f a, bf;
    a.x  = xrow[k];
    a.y  = xrow[k + 1];
    bf.x = wcol[k * CD];
    bf.y = wcol[(k + 1) * CD];
    acc = __builtin_amdgcn_wmma_f32_16x16x4_f32(
        /*neg_a=*/false, a, /*neg_b=*/false, bf,
        /*c_mod=*/(short)0, acc, /*reuse_a=*/false, /*reuse_b=*/false);
  }

  // C/D layout: VGPR r -> row (r | r+8), col = c0 + (lane & 15)
  if (cW < CD) {
    const float bv = bias[cW];
    const int rbase = row0 + ((lane < 16) ? 0 : 8);
#pragma unroll
    for (int r = 0; r < 8; ++r) {
      int rr = rbase + r;
      if (rr < n) out[(size_t)rr * CD + cW] = acc[r] + bv;
    }
  }
}

// ---------------- in-place log-softmax, one wave32 per row (C=40) ----------------

__global__ void __launch_bounds__(256)
logsoftmax_kernel(float* __restrict__ out, int n) {
  int row = blockIdx.x * 8 + (int)(threadIdx.x >> 5);
  if (row >= n) return;
  int lane = threadIdx.x & 31;
  float* p = out + (size_t)row * CD;

  float v0 = p[lane];                                  // cols 0..31
  float v1 = (lane < CD - 32) ? p[32 + lane] : -__builtin_inff();

  float mx = fmaxf(v0, v1);
#pragma unroll
  for (int off = 16; off > 0; off >>= 1)
    mx = fmaxf(mx, __shfl_xor(mx, off, 32));

  float s = __expf(v0 - mx) + ((lane < CD - 32) ? __expf(v1 - mx) : 0.0f);
#pragma unroll
  for (int off = 16; off > 0; off >>= 1)
    s += __shfl_xor(s, off, 32);

  float lse = mx + __logf(s);
  p[lane] = v0 - lse;
  if (lane < CD - 32) p[32 + lane] = v1 - lse;
}

// ---------------------------------------------------------------------------

extern "C" void kernel_launch(void* const* d_in, const int* in_sizes, int n_in,
                              void* d_out, int out_size, void* d_ws, size_t ws_size,
                              hipStream_t stream) {
  const float*     x  = (const float*)d_in[0];      // [N,64]
  const long long* ei = (const long long*)d_in[1];  // [2,E] int64
  const float*     Wm = (const float*)d_in[2];      // [64,40]
  const float*     bb = (const float*)d_in[3];      // [40]
  float* out = (float*)d_out;

  const int N_ = in_sizes[0] / DD;
  const int E_ = in_sizes[1] / 2;

  float* ws   = (float*)d_ws;
  float* deg  = ws;                                 // N floats (becomes deg_inv_sqrt)
  float* buf0 = ws + N_;                            // N*64
  float* buf1 = buf0 + (size_t)N_ * DD;             // N*64

  // 1) degree (incl. self loop) -> deg_inv_sqrt, recomputed every call
  init_deg_kernel<<<(N_ + 255) / 256, 256, 0, stream>>>(deg, N_);
  deg_edges_kernel<<<(E_ + 255) / 256, 256, 0, stream>>>(ei, deg, E_);
  rsqrt_inplace_kernel<<<(N_ + 255) / 256, 256, 0, stream>>>(deg, N_);

  // 2) hop 1: x -> buf0
  self_init_kernel<<<(N_ * 16 + 255) / 256, 256, 0, stream>>>(x, deg, buf0, N_);
  spmm_edges_kernel<<<(E_ + 7) / 8, 256, 0, stream>>>(ei, deg, x, buf0, E_);

  // 3) hop 2: buf0 -> buf1
  self_init_kernel<<<(N_ * 16 + 255) / 256, 256, 0, stream>>>(buf0, deg, buf1, N_);
  spmm_edges_kernel<<<(E_ + 7) / 8, 256, 0, stream>>>(ei, deg, buf0, buf1, E_);

  // 4) FP32 WMMA GEMM + bias -> d_out (logits)
  gemm_wmma_kernel<<<(N_ + 15) / 16, 96, 0, stream>>>(buf1, Wm, bb, out, N_);

  // 5) in-place log-softmax
  logsoftmax_kernel<<<(N_ + 7) / 8, 256, 0, stream>>>(out, N_);
}